// PrunedQwen3MoeSparseMoeBlock_48593259986978
// MI455X (gfx1250) — compile-verified
//
#include <hip/hip_runtime.h>
#include <math.h>

typedef __attribute__((ext_vector_type(2))) float v2f;
typedef __attribute__((ext_vector_type(4))) float v4f;
typedef __attribute__((ext_vector_type(8))) float v8f;

#define HID   2048   // hidden size
#define NI    768    // intermediate size
#define NE    48     // kept experts
#define CAPE  768    // capacity per expert
#define NEXP  128    // total experts (router)
#define TOKS  2048   // batch*seq
#define TKK   8      // top-k
#define KC    64     // K chunk per LDS stage
#define LDA   68     // padded LDS row (floats): 68*4=272B, 16B-aligned rows

__device__ __forceinline__ v8f wmma4(v2f a, v2f b, v8f c) {
  // V_WMMA_F32_16X16X4_F32 : D = A(16x4) x B(4x16) + C(16x16)
  return __builtin_amdgcn_wmma_f32_16x16x4_f32(false, a, false, b, (short)0, c,
                                               false, false);
}

// ---------------------------------------------------------------- zero output
__global__ __launch_bounds__(256) void zero_f32(float* __restrict__ p, int n) {
  int i = blockIdx.x * blockDim.x + threadIdx.x;
  int stride = gridDim.x * blockDim.x;
  for (; i < n; i += stride) p[i] = 0.0f;
}

// ------------------------------------------------------- router logits (WMMA)
// logits[TOKS x NEXP] = x[TOKS x HID] @ gate_weight[NEXP x HID]^T
__global__ __launch_bounds__(256) void router_gemm(
    const float* __restrict__ x, const float* __restrict__ gw,
    float* __restrict__ logits) {
  __shared__ float As[64][LDA];
  __shared__ float Bs[64][LDA];
  int m0 = blockIdx.x * 64;
  int n0 = blockIdx.y * 64;
  int tid = threadIdx.x;
  int lane = tid & 31, wv = tid >> 5;
  int msub = (wv & 3) * 16, nsub = (wv >> 2) * 16;
  int lrow = tid >> 2, lc0 = (tid & 3) * 16;
  const float* ap = x + (size_t)(m0 + lrow) * HID + lc0;
  const float* bp = gw + (size_t)(n0 + lrow) * HID + lc0;
  v8f a0 = {}, a1 = {};
  int klo = (lane >> 4) * 2;
  int am = msub + (lane & 15);
  int bn = nsub + (lane & 15);
  for (int kb = 0; kb < HID; kb += KC) {
    v4f av[4], bv[4];
#pragma unroll
    for (int q = 0; q < 4; q++) {
      av[q] = *(const v4f*)(ap + kb + 4 * q);
      bv[q] = *(const v4f*)(bp + kb + 4 * q);
    }
    __syncthreads();
#pragma unroll
    for (int q = 0; q < 4; q++) {
      *(v4f*)&As[lrow][lc0 + 4 * q] = av[q];
      *(v4f*)&Bs[lrow][lc0 + 4 * q] = bv[q];
    }
    __syncthreads();
#pragma unroll
    for (int k = 0; k < KC; k += 4) {
      v2f af = *(const v2f*)&As[am][k + klo];
      v2f b0 = *(const v2f*)&Bs[bn][k + klo];
      v2f b1 = *(const v2f*)&Bs[bn + 32][k + klo];
      a0 = wmma4(af, b0, a0);
      a1 = wmma4(af, b1, a1);
    }
  }
  int mmb = (lane >> 4) * 8;
  int col = n0 + nsub + (lane & 15);
#pragma unroll
  for (int r = 0; r < 8; r++) {
    int row = m0 + msub + mmb + r;
    logits[(size_t)row * NEXP + col] = a0[r];
    logits[(size_t)row * NEXP + col + 32] = a1[r];
  }
}

// -------------------------------------------------- per-token top-8 + softmax
// one wave per token; mask pruned experts; jax tie-break (lowest index wins)
__global__ __launch_bounds__(256) void router_topk(
    const float* __restrict__ logits, const int* __restrict__ o2n,
    int* __restrict__ topk_e, float* __restrict__ topk_w) {
  int lane = threadIdx.x & 31;
  int wv = threadIdx.x >> 5;
  int t = blockIdx.x * 8 + wv;
  const float* lg = logits + (size_t)t * NEXP;
  float v[4];
  int base = lane * 4;
#pragma unroll
  for (int j = 0; j < 4; j++) {
    int e = base + j;
    v[j] = (o2n[e] == -1) ? -INFINITY : lg[e];
  }
  float selv[TKK];
  int seli[TKK];
#pragma unroll
  for (int k = 0; k < TKK; k++) {
    float bv = v[0];
    int bi = base;
#pragma unroll
    for (int j = 1; j < 4; j++)
      if (v[j] > bv) { bv = v[j]; bi = base + j; }
#pragma unroll
    for (int off = 16; off > 0; off >>= 1) {
      float ov = __shfl_xor(bv, off, 32);
      int oi = __shfl_xor(bi, off, 32);
      if (ov > bv || (ov == bv && oi < bi)) { bv = ov; bi = oi; }
    }
    selv[k] = bv;
    seli[k] = bi;
    if ((bi >> 2) == lane) v[bi & 3] = -INFINITY;
  }
  float mx = selv[0];
  float wk[TKK];
  float s = 0.0f;
#pragma unroll
  for (int k = 0; k < TKK; k++) { wk[k] = __expf(selv[k] - mx); s += wk[k]; }
  float inv = 1.0f / s;
  if (lane < TKK) {
    topk_e[t * TKK + lane] = o2n[seli[lane]];
    topk_w[t * TKK + lane] = wk[lane] * inv;
  }
}

// ---------------------------------- exact-order capacity assignment per expert
__global__ __launch_bounds__(32) void assign_slots(
    const int* __restrict__ topk_e, const float* __restrict__ topk_w,
    int* __restrict__ slot_tok, float* __restrict__ slot_w,
    int* __restrict__ cnt) {
  int e = blockIdx.x;
  int lane = threadIdx.x;
  int count = 0;
  for (int b = 0; b < TOKS * TKK; b += 32) {
    int i = b + lane;
    bool m = (topk_e[i] == e);
    unsigned mask = (unsigned)__ballot(m);
    int prefix = __popc(mask & ((1u << lane) - 1u));
    if (m) {
      int pos = count + prefix;
      if (pos < CAPE) {
        slot_tok[e * CAPE + pos] = i >> 3;  // token id
        slot_w[e * CAPE + pos] = topk_w[i];
      }
    }
    count += __popc(mask);
  }
  if (lane == 0) cnt[e] = (count < CAPE) ? count : CAPE;
}

// ------------------- GEMM1: gathered x @ gate_up^T, fused silu(g)*u -> hbuf
// weights streamed with non-temporal loads (one-pass, 604MB) so the re-used
// activation working set (x, 16MB) stays L2-resident.
__global__ __launch_bounds__(256) void moe_gemm1(
    const float* __restrict__ x, const float* __restrict__ wgu,
    const int* __restrict__ slot_tok, const int* __restrict__ cnt,
    float* __restrict__ hbuf) {
  int e = blockIdx.z;
  int c = cnt[e];
  int m0 = blockIdx.x * 64;
  if (m0 >= c) return;
  int n0 = blockIdx.y * 32;
  __shared__ float As[64][LDA];
  __shared__ float Bg[32][LDA];
  __shared__ float Bu[32][LDA];
  int tid = threadIdx.x, lane = tid & 31, wv = tid >> 5;
  int msub = (wv & 3) * 16, nsub = (wv >> 2) * 16;
  int arow = tid >> 2, ac0 = (tid & 3) * 16;
  int gslot = m0 + arow;
  const float* ap = nullptr;
  if (gslot < c) ap = x + (size_t)slot_tok[e * CAPE + gslot] * HID + ac0;
  int brow = tid >> 3, bc0 = (tid & 7) * 8;
  const float* gp = wgu + ((size_t)e * (2 * NI) + n0 + brow) * HID + bc0;
  const float* up = gp + (size_t)NI * HID;
  v8f ag = {}, au = {};
  int klo = (lane >> 4) * 2;
  int am = msub + (lane & 15);
  int bn = nsub + (lane & 15);
  for (int kb = 0; kb < HID; kb += KC) {
    v4f av[4] = {};
    if (ap) {
#pragma unroll
      for (int q = 0; q < 4; q++) av[q] = *(const v4f*)(ap + kb + 4 * q);
    }
    v4f gv[2], uv[2];
#pragma unroll
    for (int q = 0; q < 2; q++) {
      gv[q] = __builtin_nontemporal_load((const v4f*)(gp + kb + 4 * q));
      uv[q] = __builtin_nontemporal_load((const v4f*)(up + kb + 4 * q));
    }
    __syncthreads();
#pragma unroll
    for (int q = 0; q < 4; q++) *(v4f*)&As[arow][ac0 + 4 * q] = av[q];
#pragma unroll
    for (int q = 0; q < 2; q++) {
      *(v4f*)&Bg[brow][bc0 + 4 * q] = gv[q];
      *(v4f*)&Bu[brow][bc0 + 4 * q] = uv[q];
    }
    __syncthreads();
#pragma unroll
    for (int k = 0; k < KC; k += 4) {
      v2f af = *(const v2f*)&As[am][k + klo];
      v2f bg = *(const v2f*)&Bg[bn][k + klo];
      v2f bu = *(const v2f*)&Bu[bn][k + klo];
      ag = wmma4(af, bg, ag);
      au = wmma4(af, bu, au);
    }
  }
  int mmb = (lane >> 4) * 8;
  int colo = n0 + nsub + (lane & 15);
#pragma unroll
  for (int r = 0; r < 8; r++) {
    int row = m0 + msub + mmb + r;
    float g = ag[r], u = au[r];
    float hv = (g / (1.0f + __expf(-g))) * u;  // silu(g) * u
    hbuf[((size_t)e * CAPE + row) * NI + colo] = hv;
  }
}

// ------------------------ GEMM2: h @ down^T, weighted atomic scatter to out
__global__ __launch_bounds__(256) void moe_gemm2(
    const float* __restrict__ hbuf, const float* __restrict__ wd,
    const int* __restrict__ slot_tok, const float* __restrict__ slot_w,
    const int* __restrict__ cnt, float* __restrict__ out) {
  int e = blockIdx.z;
  int c = cnt[e];
  int m0 = blockIdx.x * 64;
  if (m0 >= c) return;
  int n0 = blockIdx.y * 64;
  __shared__ float As[64][LDA];
  __shared__ float Bs[64][LDA];
  int tid = threadIdx.x, lane = tid & 31, wv = tid >> 5;
  int msub = (wv & 3) * 16, nsub = (wv >> 2) * 16;
  int lrow = tid >> 2, lc0 = (tid & 3) * 16;
  const float* ap = hbuf + ((size_t)e * CAPE + m0 + lrow) * NI + lc0;
  const float* bp = wd + ((size_t)e * HID + n0 + lrow) * NI + lc0;
  v8f a0 = {}, a1 = {};
  int klo = (lane >> 4) * 2;
  int am = msub + (lane & 15);
  int bn = nsub + (lane & 15);
  for (int kb = 0; kb < NI; kb += KC) {
    v4f av[4], bv[4];
#pragma unroll
    for (int q = 0; q < 4; q++) {
      av[q] = *(const v4f*)(ap + kb + 4 * q);
      bv[q] = __builtin_nontemporal_load((const v4f*)(bp + kb + 4 * q));
    }
    __syncthreads();
#pragma unroll
    for (int q = 0; q < 4; q++) {
      *(v4f*)&As[lrow][lc0 + 4 * q] = av[q];
      *(v4f*)&Bs[lrow][lc0 + 4 * q] = bv[q];
    }
    __syncthreads();
#pragma unroll
    for (int k = 0; k < KC; k += 4) {
      v2f af = *(const v2f*)&As[am][k + klo];
      v2f b0 = *(const v2f*)&Bs[bn][k + klo];
      v2f b1 = *(const v2f*)&Bs[bn + 32][k + klo];
      a0 = wmma4(af, b0, a0);
      a1 = wmma4(af, b1, a1);
    }
  }
  int mmb = (lane >> 4) * 8;
  int col = n0 + nsub + (lane & 15);
#pragma unroll
  for (int r = 0; r < 8; r++) {
    int slot = m0 + msub + mmb + r;
    if (slot < c) {
      int tok = slot_tok[e * CAPE + slot];
      float w = slot_w[e * CAPE + slot];
      float* o = out + (size_t)tok * HID + col;
      __hip_atomic_fetch_add(o, a0[r] * w, __ATOMIC_RELAXED,
                             __HIP_MEMORY_SCOPE_AGENT);
      __hip_atomic_fetch_add(o + 32, a1[r] * w, __ATOMIC_RELAXED,
                             __HIP_MEMORY_SCOPE_AGENT);
    }
  }
}

// ---------------------------------------------------------------------- launch
extern "C" void kernel_launch(void* const* d_in, const int* in_sizes, int n_in,
                              void* d_out, int out_size, void* d_ws,
                              size_t ws_size, hipStream_t stream) {
  const float* x   = (const float*)d_in[0];  // hidden_states (2,1024,2048)
  const float* gw  = (const float*)d_in[1];  // gate_weight (128,2048)
  const float* wgu = (const float*)d_in[2];  // gate_up_proj (48,1536,2048)
  const float* wd  = (const float*)d_in[3];  // down_proj (48,2048,768)
  const int* o2n   = (const int*)d_in[4];    // old_to_new (128,)
  float* out = (float*)d_out;

  char* ws = (char*)d_ws;
  size_t off = 0;
  auto carve = [&](size_t bytes) -> char* {
    char* p = ws + off;
    off += (bytes + 255) & ~(size_t)255;
    return p;
  };
  float* logits  = (float*)carve((size_t)TOKS * NEXP * 4);
  int*   topk_e  = (int*)carve((size_t)TOKS * TKK * 4);
  float* topk_w  = (float*)carve((size_t)TOKS * TKK * 4);
  int*   slot_t  = (int*)carve((size_t)NE * CAPE * 4);
  float* slot_w  = (float*)carve((size_t)NE * CAPE * 4);
  int*   cnt     = (int*)carve((size_t)NE * 4);
  float* hbuf    = (float*)carve((size_t)NE * CAPE * NI * 4);  // ~113 MB

  zero_f32<<<2048, 256, 0, stream>>>(out, TOKS * HID);
  router_gemm<<<dim3(TOKS / 64, NEXP / 64), 256, 0, stream>>>(x, gw, logits);
  router_topk<<<TOKS / 8, 256, 0, stream>>>(logits, o2n, topk_e, topk_w);
  assign_slots<<<NE, 32, 0, stream>>>(topk_e, topk_w, slot_t, slot_w, cnt);
  moe_gemm1<<<dim3(CAPE / 64, NI / 32, NE), 256, 0, stream>>>(x, wgu, slot_t,
                                                              cnt, hbuf);
  moe_gemm2<<<dim3(CAPE / 64, HID / 64, NE), 256, 0, stream>>>(
      hbuf, wd, slot_t, slot_w, cnt, out);
}